// CTCLayer_39298950759021
// MI455X (gfx1250) — compile-verified
//
#include <hip/hip_runtime.h>
#include <math.h>

// ---------------------------------------------------------------------------
// CTC forward (alpha) recursion, Keras convention (blank = C-1).
// B=256 samples, T=512 steps, C=128 classes, L=48 labels, S=2L+1=97 states.
// One block per sample, 128 threads (4 wave32s).
// CDNA5 path: async global->LDS staging of probability rows, triple-buffered,
// synchronized with s_wait_asynccnt + workgroup split-barriers.
// ---------------------------------------------------------------------------

#ifndef __has_builtin
#define __has_builtin(x) 0
#endif

#if defined(__HIP_DEVICE_COMPILE__) && \
    __has_builtin(__builtin_amdgcn_global_load_async_to_lds_b32) && \
    __has_builtin(__builtin_amdgcn_s_wait_asynccnt)
#define CTC_HAS_ASYNC 1
#else
#define CTC_HAS_ASYNC 0
#endif

#define CTC_GLOBAL_AS __attribute__((address_space(1)))
#define CTC_LDS_AS    __attribute__((address_space(3)))

__device__ __forceinline__ void async_copy_f32(const float* g, float* l) {
#if CTC_HAS_ASYNC
  // global_load_async_to_lds_b32: per-lane 4B DMA, tracked with ASYNCcnt.
  // Builtin signature (from clang diagnostic): (AS1 int*, AS3 int*, imm, imm).
  __builtin_amdgcn_global_load_async_to_lds_b32(
      (CTC_GLOBAL_AS int*)g, (CTC_LDS_AS int*)l, 0, 0);
#else
  *l = *g;  // synchronous fallback (still correct)
#endif
}

template <int N>
__device__ __forceinline__ void wait_async() {
#if CTC_HAS_ASYNC
  __builtin_amdgcn_s_wait_asynccnt(N);
#endif
}

namespace {
constexpr int   kT     = 512;
constexpr int   kC     = 128;
constexpr int   kL     = 48;
constexpr int   kS     = 2 * kL + 1;   // 97 extended states
constexpr int   kBlank = kC - 1;
constexpr float kNegInf = -1e30f;
constexpr float kEps    = 1e-7f;
}  // namespace

__global__ __launch_bounds__(128) void ctc_fwd_kernel(
    const int* __restrict__ labels,   // [B, L] int32
    const float* __restrict__ probs,  // [B, T, C] softmax probabilities
    float* __restrict__ loss)         // [B] (viewed as [B,1])
{
  __shared__ float rowbuf[3][kC];        // triple-buffered prob rows
  __shared__ float alpha[2][kS + 2];     // ping-pong alpha, 2-slot -inf pad

  const int b   = blockIdx.x;
  const int tid = threadIdx.x;           // 0..127: state id (if <97) & class id
  const float* base = probs + (size_t)b * (size_t)kT * (size_t)kC;

  // Kick off async staging of rows t=0 and t=1 (one float per lane).
  async_copy_f32(base + tid,       &rowbuf[0][tid]);
  async_copy_f32(base + kC + tid,  &rowbuf[1][tid]);

  // Initialize both alpha buffers (incl. the 2-slot front padding) to -inf.
  if (tid < kS + 2) {
    alpha[0][tid] = kNegInf;
    alpha[1][tid] = kNegInf;
  }

  // Per-state metadata: class of extended state, and skip-transition legality.
  // ext[2k] = blank, ext[2k+1] = label[k]; skip allowed iff state is a label
  // state and its label differs from the label two states back (blank-padded).
  int cls = kBlank;
  int canskip = 0;
  if ((tid < kS) && (tid & 1)) {
    const int k = tid >> 1;
    cls = labels[b * kL + k];
    const int prev = (tid >= 3) ? labels[b * kL + k - 1] : kBlank;
    canskip = (cls != kBlank) && (cls != prev);
  }

  // label_len = count_nonzero(labels) -> end state index (thread 0 only).
  int end_idx = 0;
  if (tid == 0) {
    int cnt = 0;
    for (int k = 0; k < kL; ++k) cnt += (labels[b * kL + k] != 0) ? 1 : 0;
    end_idx = 2 * cnt;
  }

  wait_async<1>();   // this wave's lanes of row 0 have landed in LDS
  __syncthreads();   // all waves: row 0 resident + alpha init visible

  // t = 0: alpha[0] = lp[0][blank-state 0], alpha[1] = lp[0][label-state 1].
  if (tid <= 1) {
    const float p = rowbuf[0][(tid == 0) ? kBlank : cls];
    alpha[0][2 + tid] = __logf(p + kEps);
  }

  int cur = 0;
  for (int t = 1; t < kT; ++t) {
    const int rb = t % 3;
    if (t + 1 < kT) {
      // Prefetch row t+1; buffer (t+1)%3 was last read 2 barriers ago -> safe.
      async_copy_f32(base + (size_t)(t + 1) * kC + tid,
                     &rowbuf[(t + 1) % 3][tid]);
      wait_async<1>();   // in-order completion => row t is done
    } else {
      wait_async<0>();   // drain the final row
    }
    __syncthreads();     // row t visible from all waves; alpha[cur] complete

    if (tid < kS) {
      const float* a = &alpha[cur][2];
      const float a0 = a[tid];
      const float a1 = a[tid - 1];                       // pad gives -inf @ s=0
      const float a2 = canskip ? a[tid - 2] : kNegInf;   // pad gives -inf @ s=1
      const float m  = fmaxf(fmaxf(a0, a1), a2);
      const float s  = __expf(a0 - m) + __expf(a1 - m) + __expf(a2 - m);
      const float lse = m + __logf(s);
      alpha[cur ^ 1][2 + tid] = lse + __logf(rowbuf[rb][cls] + kEps);
    }
    cur ^= 1;
  }

  __syncthreads();
  if (tid == 0) {
    const float ae = alpha[cur][2 + end_idx];
    const float ap = alpha[cur][2 + end_idx - 1];   // end_idx==0 -> pad -inf
    const float m  = fmaxf(ae, ap);
    loss[b] = -(m + __logf(__expf(ae - m) + __expf(ap - m)));
  }
}

extern "C" void kernel_launch(void* const* d_in, const int* in_sizes, int n_in,
                              void* d_out, int out_size, void* d_ws,
                              size_t ws_size, hipStream_t stream) {
  (void)n_in; (void)out_size; (void)d_ws; (void)ws_size;
  const int*   y_true = (const int*)d_in[0];    // [B, L] int32
  const float* y_pred = (const float*)d_in[1];  // [B, T, C] float32
  float*       out    = (float*)d_out;          // [B, 1] float32

  const int B = in_sizes[0] / kL;               // 256
  hipLaunchKernelGGL(ctc_fwd_kernel, dim3(B), dim3(128), 0, stream,
                     y_true, y_pred, out);
}